// GCN_3633542333206
// MI455X (gfx1250) — compile-verified
//
#include <hip/hip_runtime.h>
#include <hip/hip_bf16.h>
#include <math.h>

// ---------------------------------------------------------------------------
// GCN forward for MI455X (gfx1250, wave32).
// GEMMs use V_WMMA_F32_16X16X4_F32 (exact f32 matrix math; GEMM is not the
// bottleneck -- the edge scatter/gather is, and it is L2-resident: the
// 100k x 128 f32 feature matrix is 51 MB vs 192 MB global L2).
// ---------------------------------------------------------------------------

typedef __attribute__((ext_vector_type(2))) float v2f;
typedef __attribute__((ext_vector_type(8))) float v8f;

#define GK      128        // inner dim (D_IN == D_H == 128)
#define NT      4          // 4 column tiles of 16 -> 64 cols per block.y
#define LDK     130        // LDS stride (floats) for transposed W: conflict-free b64
#define DOUT    47
#define BN_EPS  1e-5f
#define BN_ROWS 512        // rows per BN partial block

// ---------------------------------------------------------------------------
// Degree / normalization
// ---------------------------------------------------------------------------
__global__ void k_deg_init(float* deg, int n) {
    int i = blockIdx.x * blockDim.x + threadIdx.x;
    if (i < n) deg[i] = 1.0f;                 // self-loop contribution
}

__global__ void k_deg_count(const int* __restrict__ dst, float* deg, int ne) {
    int i = blockIdx.x * blockDim.x + threadIdx.x;
    if (i < ne) atomicAdd(&deg[dst[i]], 1.0f);
}

__global__ void k_rsqrt_inplace(float* deg, int n) {
    int i = blockIdx.x * blockDim.x + threadIdx.x;
    if (i < n) deg[i] = rsqrtf(deg[i]);       // deg >= 1 always
}

// ---------------------------------------------------------------------------
// WMMA GEMM: Y[nrows x ncols] = X[nrows x 128] * W[128 x ncols]
// block = 256 threads (8 waves); wave owns a 16-row x 64-col strip.
// grid = ((nrows+127)/128, (ncols+63)/64)
//
// OOB rows: A-row M only contributes to D-row M, and OOB D rows are never
// stored, so we simply clamp the A row index and load unconditionally --
// keeps the inner loop branch-free (no per-step exec save/restore).
// ---------------------------------------------------------------------------
__global__ void k_gemm_wmma(const float* __restrict__ X,
                            const float* __restrict__ W,
                            float* __restrict__ Y,
                            int nrows, int ncols) {
    __shared__ float lw[NT * 16 * LDK];       // W^T tile: [col][k], padded stride

    const int tid     = threadIdx.x;
    const int colbase = blockIdx.y * (NT * 16);

    // Stage W^T into LDS (zero-pad out-of-range columns).
    for (int idx = tid; idx < GK * NT * 16; idx += blockDim.x) {
        int k  = idx / (NT * 16);
        int c  = idx - k * (NT * 16);
        int gc = colbase + c;
        lw[c * LDK + k] = (gc < ncols) ? W[(size_t)k * ncols + gc] : 0.0f;
    }
    __syncthreads();

    const int wave    = tid >> 5;
    const int lane    = tid & 31;
    const int half    = lane >> 4;            // 0: K=k0,k0+1   1: K=k0+2,k0+3
    const int m       = lane & 15;
    const int rowbase = blockIdx.x * 128 + wave * 16;
    const int arow    = min(rowbase + m, nrows - 1);   // clamped, always valid
    const float2* xrow = (const float2*)(X + (size_t)arow * GK);
    const float* lwm   = &lw[m * LDK];

    v8f acc[NT];
#pragma unroll
    for (int t = 0; t < NT; ++t)
#pragma unroll
        for (int r = 0; r < 8; ++r) acc[t][r] = 0.0f;

    for (int k0 = 0; k0 < GK; k0 += 4) {
        const int kk = k0 + half * 2;
        const float2 t2 = xrow[kk >> 1];      // global_load_b64, 8B aligned
        v2f a; a.x = t2.x; a.y = t2.y;
#pragma unroll
        for (int t = 0; t < NT; ++t) {
            // B fragment: B[kk][col], B[kk+1][col] -> consecutive in W^T LDS
            const float2 bb = *(const float2*)&lwm[t * 16 * LDK + kk];
            v2f b; b.x = bb.x; b.y = bb.y;
            acc[t] = __builtin_amdgcn_wmma_f32_16x16x4_f32(
                false, a, false, b, (short)0, acc[t], false, false);
        }
    }

    // D layout: VGPR r -> row = rowbase + half*8 + r, col = tile*16 + m
#pragma unroll
    for (int t = 0; t < NT; ++t) {
        const int cc = colbase + t * 16 + m;
        if (cc < ncols) {
#pragma unroll
            for (int r = 0; r < 8; ++r) {
                const int rr = rowbase + half * 8 + r;
                if (rr < nrows) Y[(size_t)rr * ncols + cc] = acc[t][r];
            }
        }
    }
}

// ---------------------------------------------------------------------------
// Aggregation: out = h*dinv^2 + bias (self-loop), then atomic edge scatter.
// C=128 paths are float4-vectorized (b128 loads, 4 f32 atomics per lane).
// ---------------------------------------------------------------------------
__global__ void k_selfloop_bias128(const float* __restrict__ h,
                                   const float* __restrict__ dinv,
                                   const float* __restrict__ bias,
                                   float* __restrict__ out, int nrows) {
    int i = blockIdx.x * blockDim.x + threadIdx.x;   // one float4 group
    if (i < nrows * 32) {
        int r  = i >> 5;
        int c4 = (i & 31) * 4;
        float di = dinv[r];
        float dd = di * di;
        float4 hv = *(const float4*)(h + (size_t)r * 128 + c4);
        float4 bv = *(const float4*)(bias + c4);
        float4 o;
        o.x = hv.x * dd + bv.x;
        o.y = hv.y * dd + bv.y;
        o.z = hv.z * dd + bv.z;
        o.w = hv.w * dd + bv.w;
        *(float4*)(out + (size_t)r * 128 + c4) = o;
    }
}

__global__ void k_scatter128(const float* __restrict__ h,
                             const int* __restrict__ src,
                             const int* __restrict__ dst,
                             const float* __restrict__ dinv,
                             float* __restrict__ out, int nedges) {
    const int lane = threadIdx.x & 31;
    const int wid  = (blockIdx.x * blockDim.x + threadIdx.x) >> 5;
    const int nw   = (gridDim.x * blockDim.x) >> 5;
    for (int e = wid; e < nedges; e += nw) {
        const int s = src[e];
        const int d = dst[e];
        const float w = dinv[s] * dinv[d];
        const float4 v = *(const float4*)(h + (size_t)s * 128 + lane * 4);
        float* od = out + (size_t)d * 128 + lane * 4;
        atomicAdd(od + 0, v.x * w);
        atomicAdd(od + 1, v.y * w);
        atomicAdd(od + 2, v.z * w);
        atomicAdd(od + 3, v.w * w);
    }
}

// Scalar variants for the 47-class layer
__global__ void k_selfloop_bias47(const float* __restrict__ h,
                                  const float* __restrict__ dinv,
                                  const float* __restrict__ bias,
                                  float* __restrict__ out, int nrows) {
    int i = blockIdx.x * blockDim.x + threadIdx.x;
    if (i < nrows * DOUT) {
        int r = i / DOUT;
        int c = i - r * DOUT;
        float di = dinv[r];
        out[i] = h[i] * di * di + bias[c];
    }
}

__global__ void k_scatter47(const float* __restrict__ h,
                            const int* __restrict__ src,
                            const int* __restrict__ dst,
                            const float* __restrict__ dinv,
                            float* __restrict__ out, int nedges) {
    const int lane = threadIdx.x & 31;
    const int wid  = (blockIdx.x * blockDim.x + threadIdx.x) >> 5;
    const int nw   = (gridDim.x * blockDim.x) >> 5;
    for (int e = wid; e < nedges; e += nw) {
        const int s = src[e];
        const int d = dst[e];
        const float w = dinv[s] * dinv[d];
        const float* hs = h + (size_t)s * DOUT;
        float* od = out + (size_t)d * DOUT;
        if (lane < DOUT)      atomicAdd(&od[lane],      hs[lane]      * w);
        if (lane + 32 < DOUT) atomicAdd(&od[lane + 32], hs[lane + 32] * w);
    }
}

// ---------------------------------------------------------------------------
// BatchNorm (deterministic two-stage reduction) + fused affine/ReLU
// ---------------------------------------------------------------------------
__global__ void k_bn_partial(const float* __restrict__ h,
                             float* __restrict__ partials, int nrows) {
    const int c  = threadIdx.x;               // 128 threads = 128 channels
    const int r0 = blockIdx.x * BN_ROWS;
    const int r1 = min(r0 + BN_ROWS, nrows);
    float s = 0.0f, q = 0.0f;
    for (int r = r0; r < r1; ++r) {
        float v = h[(size_t)r * 128 + c];
        s += v; q += v * v;
    }
    partials[blockIdx.x * 256 + c]       = s;
    partials[blockIdx.x * 256 + 128 + c] = q;
}

__global__ void k_bn_finalize(const float* __restrict__ partials, int nblk,
                              const float* __restrict__ g,
                              const float* __restrict__ be,
                              float* __restrict__ ss, int nrows) {
    const int c = threadIdx.x;                // 128 threads
    float s = 0.0f, q = 0.0f;
    for (int b = 0; b < nblk; ++b) {
        s += partials[b * 256 + c];
        q += partials[b * 256 + 128 + c];
    }
    const float inv_n = 1.0f / (float)nrows;
    const float mean  = s * inv_n;
    const float var   = q * inv_n - mean * mean;
    const float sc    = g[c] * rsqrtf(var + BN_EPS);
    ss[c]       = sc;
    ss[128 + c] = be[c] - mean * sc;
}

__global__ void k_bn_relu4(float* __restrict__ h,
                           const float* __restrict__ ss, int ngroups) {
    int i = blockIdx.x * blockDim.x + threadIdx.x;   // one float4 group
    if (i < ngroups) {
        int c4 = (i & 31) * 4;
        float4 v  = *(float4*)(h + (size_t)i * 4);
        float4 sc = *(const float4*)(ss + c4);
        float4 sh = *(const float4*)(ss + 128 + c4);
        v.x = fmaxf(v.x * sc.x + sh.x, 0.0f);
        v.y = fmaxf(v.y * sc.y + sh.y, 0.0f);
        v.z = fmaxf(v.z * sc.z + sh.z, 0.0f);
        v.w = fmaxf(v.w * sc.w + sh.w, 0.0f);
        *(float4*)(h + (size_t)i * 4) = v;
    }
}

// ---------------------------------------------------------------------------
// In-place row log-softmax over DOUT=47 classes; one wave per row.
// ---------------------------------------------------------------------------
__global__ void k_log_softmax(float* __restrict__ out, int nrows) {
    const int lane = threadIdx.x & 31;
    const int row  = (blockIdx.x * blockDim.x + threadIdx.x) >> 5;
    if (row >= nrows) return;
    float* p = out + (size_t)row * DOUT;

    const float NEG_INF = -__builtin_inff();
    float v0 = (lane < DOUT)      ? p[lane]      : NEG_INF;
    float v1 = (lane + 32 < DOUT) ? p[lane + 32] : NEG_INF;

    float mx = fmaxf(v0, v1);
#pragma unroll
    for (int o = 16; o > 0; o >>= 1) mx = fmaxf(mx, __shfl_xor(mx, o, 32));

    float e = ((lane < DOUT) ? expf(v0 - mx) : 0.0f) +
              ((lane + 32 < DOUT) ? expf(v1 - mx) : 0.0f);
#pragma unroll
    for (int o = 16; o > 0; o >>= 1) e += __shfl_xor(e, o, 32);

    const float lse = mx + logf(e);
    if (lane < DOUT)      p[lane]      = v0 - lse;
    if (lane + 32 < DOUT) p[lane + 32] = v1 - lse;
}

// ---------------------------------------------------------------------------
// Launch sequence
// ---------------------------------------------------------------------------
extern "C" void kernel_launch(void* const* d_in, const int* in_sizes, int n_in,
                              void* d_out, int out_size, void* d_ws, size_t ws_size,
                              hipStream_t stream) {
    const float* x   = (const float*)d_in[0];
    const int*   ei  = (const int*)  d_in[1];
    const float* W0  = (const float*)d_in[2];
    const float* b0  = (const float*)d_in[3];
    const float* Wm  = (const float*)d_in[4];
    const float* bm  = (const float*)d_in[5];
    const float* Wl  = (const float*)d_in[6];
    const float* bl  = (const float*)d_in[7];
    const float* g0  = (const float*)d_in[8];
    const float* be0 = (const float*)d_in[9];
    const float* g1  = (const float*)d_in[10];
    const float* be1 = (const float*)d_in[11];

    const int nrows  = in_sizes[0] / GK;         // 100000
    const int nedges = in_sizes[1] / 2;          // 1600000
    const int* src   = ei;
    const int* dst   = ei + nedges;

    // Workspace carve-up (floats)
    float* ws   = (float*)d_ws;
    float* dinv = ws;                                       // [nrows]
    size_t off  = ((size_t)nrows + 255) & ~(size_t)255;
    float* bufA = ws + off;                                 // [nrows*128]
    float* bufB = bufA + (size_t)nrows * 128;               // [nrows*128]
    float* ss   = bufB + (size_t)nrows * 128;               // [256] BN scale/shift
    float* part = ss + 256;                                 // [nblkBN*256]

    float* outp = (float*)d_out;                            // [nrows*47]

    const int nblkBN = (nrows + BN_ROWS - 1) / BN_ROWS;
    const dim3 gemm_blk(256);
    const dim3 gemm_grid128((nrows + 127) / 128, (GK + NT * 16 - 1) / (NT * 16));
    const dim3 gemm_grid47((nrows + 127) / 128, 1);
    const int scat_blocks = 1024;
    const int ew_groups   = nrows * 32;                     // float4 groups
    const int ew_blocks   = (ew_groups + 255) / 256;

    // ---- normalization coefficients ----
    k_deg_init<<<(nrows + 255) / 256, 256, 0, stream>>>(dinv, nrows);
    k_deg_count<<<(nedges + 255) / 256, 256, 0, stream>>>(dst, dinv, nedges);
    k_rsqrt_inplace<<<(nrows + 255) / 256, 256, 0, stream>>>(dinv, nrows);

    // ---- layer 0: conv -> BN -> ReLU ----
    k_gemm_wmma<<<gemm_grid128, gemm_blk, 0, stream>>>(x, W0, bufA, nrows, GK);
    k_selfloop_bias128<<<ew_blocks, 256, 0, stream>>>(bufA, dinv, b0, bufB, nrows);
    k_scatter128<<<scat_blocks, 256, 0, stream>>>(bufA, src, dst, dinv, bufB, nedges);
    k_bn_partial<<<nblkBN, 128, 0, stream>>>(bufB, part, nrows);
    k_bn_finalize<<<1, 128, 0, stream>>>(part, nblkBN, g0, be0, ss, nrows);
    k_bn_relu4<<<ew_blocks, 256, 0, stream>>>(bufB, ss, ew_groups);

    // ---- mid layer: conv -> BN -> ReLU ----
    k_gemm_wmma<<<gemm_grid128, gemm_blk, 0, stream>>>(bufB, Wm, bufA, nrows, GK);
    k_selfloop_bias128<<<ew_blocks, 256, 0, stream>>>(bufA, dinv, bm, bufB, nrows);
    k_scatter128<<<scat_blocks, 256, 0, stream>>>(bufA, src, dst, dinv, bufB, nedges);
    k_bn_partial<<<nblkBN, 128, 0, stream>>>(bufB, part, nrows);
    k_bn_finalize<<<1, 128, 0, stream>>>(part, nblkBN, g1, be1, ss, nrows);
    k_bn_relu4<<<ew_blocks, 256, 0, stream>>>(bufB, ss, ew_groups);

    // ---- last conv (47 classes) + log_softmax ----
    k_gemm_wmma<<<gemm_grid47, gemm_blk, 0, stream>>>(bufB, Wl, bufA, nrows, DOUT);
    k_selfloop_bias47<<<((size_t)nrows * DOUT + 255) / 256, 256, 0, stream>>>(
        bufA, dinv, bl, outp, nrows);
    k_scatter47<<<scat_blocks, 256, 0, stream>>>(bufA, src, dst, dinv, outp, nedges);
    k_log_softmax<<<(nrows * 32 + 255) / 256, 256, 0, stream>>>(outp, nrows);
}